// DiscretizedGaussian_23905787969963
// MI455X (gfx1250) — compile-verified
//
#include <hip/hip_runtime.h>
#include <math.h>

typedef __attribute__((ext_vector_type(2))) float v2f;
typedef __attribute__((ext_vector_type(4))) float v4f;
typedef __attribute__((ext_vector_type(8))) float v8f;

#define CHUNKS 64
#define TPB 256

// ---------------------------------------------------------------------------
// tanh: use gfx1250 native v_tanh_f32 if the builtin exists, else exp-based.
// tanh(a) = 1 - 2/(exp(2a)+1)  (monotone, saturates correctly at +-1 in f32)
// ---------------------------------------------------------------------------
__device__ __forceinline__ float fast_tanh(float a) {
#if __has_builtin(__builtin_amdgcn_tanhf)
    return __builtin_amdgcn_tanhf(a);
#elif __has_builtin(__builtin_amdgcn_tanh_f32)
    return __builtin_amdgcn_tanh_f32(a);
#else
    float e = __expf(2.0f * a);
    return 1.0f - 2.0f / (e + 1.0f);
#endif
}

// tanh( sqrt(2/pi) * (v + 0.044715 v^3) )
__device__ __forceinline__ float gelu_cdf_tanh(float v) {
    float a = 0.7978845608028654f * fmaf(0.044715f * v, v * v, v);
    return fast_tanh(a);
}

// log prob of one element (fused gather-style discretized gaussian NLL)
__device__ __forceinline__ float elem_logprob(float m, float lv, float xv) {
    // jnp.round == round-half-to-even == rintf under default RTE
    float idx     = rintf(((xv + 1.0f) * 0.5f) * 255.0f);          // 0..255
    // x_sel = (2*idx+1)/256 - 1  (exact: odd/256 grid, fma of exact terms)
    float x_sel   = fmaf(idx, 0.0078125f, -0.99609375f);
    float inv_std = __expf(-lv);
    float centered = m - x_sel;
    const float hw = 1.0f / 255.0f;
    float tp = gelu_cdf_tanh(inv_std * (centered + hw));
    float tm = gelu_cdf_tanh(inv_std * (centered - hw));
    float diff = 0.5f * (tp - tm);                                  // cdf_plus - cdf_min
    return __logf(fmaxf(diff, 1e-10f));
}

// ---------------------------------------------------------------------------
// Kernel 1: grid = (CHUNKS, B). Each block streams n_f4/CHUNKS float4 triples,
// accumulates per-thread, reduces wave via V_WMMA_F32_16X16X4_F32 ones-trick
// (layout-robust: sum over all 32 lanes of D-vgpr0 == 2 * wave_sum), then LDS
// across the 8 waves. One f32 partial per block -> ws[b*CHUNKS + chunk].
// ---------------------------------------------------------------------------
__global__ __launch_bounds__(TPB) void dg_partial_kernel(
    const float* __restrict__ mean, const float* __restrict__ logvar,
    const float* __restrict__ x, float* __restrict__ ws, long long n_f4_batch)
{
    const int chunk = blockIdx.x;
    const int b     = blockIdx.y;
    const int t     = threadIdx.x;

    const long long chunk_f4 = (n_f4_batch + CHUNKS - 1) / (long long)CHUNKS;
    long long end = (long long)(chunk + 1) * chunk_f4;
    if (end > n_f4_batch) end = n_f4_batch;

    const v4f* m4 = (const v4f*)mean   + (long long)b * n_f4_batch;
    const v4f* l4 = (const v4f*)logvar + (long long)b * n_f4_batch;
    const v4f* x4 = (const v4f*)x      + (long long)b * n_f4_batch;

    float acc = 0.0f;
    for (long long f4 = (long long)chunk * chunk_f4 + t; f4 < end; f4 += TPB) {
        v4f mv = __builtin_nontemporal_load(&m4[f4]);   // streamed once: NT b128
        v4f lv = __builtin_nontemporal_load(&l4[f4]);
        v4f xv = __builtin_nontemporal_load(&x4[f4]);
        acc += elem_logprob(mv[0], lv[0], xv[0]);
        acc += elem_logprob(mv[1], lv[1], xv[1]);
        acc += elem_logprob(mv[2], lv[2], xv[2]);
        acc += elem_logprob(mv[3], lv[3], xv[3]);
    }

    // --- wave reduction via WMMA ones-matrix (EXEC is full here) ---
    // D = A(16x4, all ones) x B(4x16) + 0  =>  D[m][n] = sum_k B[k][n].
    // B vgpr0 = acc, vgpr1 = 0. Rows of D are identical, and D-vgpr0 holds two
    // (equal) rows across the 32 lanes, so sum_{lanes}(d[0]) = 2 * sum(acc).
    v2f a_ones; a_ones[0] = 1.0f; a_ones[1] = 1.0f;
    v2f b_in;   b_in[0]   = acc;  b_in[1]   = 0.0f;
    v8f c_zero = {};
    v8f d = __builtin_amdgcn_wmma_f32_16x16x4_f32(
        /*neg_a=*/false, a_ones, /*neg_b=*/false, b_in,
        /*c_mod=*/(short)0, c_zero, /*reuse_a=*/false, /*reuse_b=*/false);

    float r = d[0];
    #pragma unroll
    for (int m = 16; m >= 1; m >>= 1) r += __shfl_xor(r, m, 32);
    float wave_sum = 0.5f * r;

    // --- cross-wave reduction (deterministic fixed order) ---
    __shared__ float wsum[TPB / 32];
    const int lane = t & 31, wid = t >> 5;
    if (lane == 0) wsum[wid] = wave_sum;
    __syncthreads();
    if (t == 0) {
        float s = 0.0f;
        #pragma unroll
        for (int i = 0; i < TPB / 32; ++i) s += wsum[i];
        ws[(long long)b * CHUNKS + chunk] = s;
    }
}

// ---------------------------------------------------------------------------
// Kernel 2: one wave per batch reduces its CHUNKS partials (fixed butterfly).
// ---------------------------------------------------------------------------
__global__ __launch_bounds__(32) void dg_final_kernel(
    const float* __restrict__ ws, float* __restrict__ out)
{
    const int b = blockIdx.x;
    const int lane = threadIdx.x;
    float v = ws[(long long)b * CHUNKS + lane] +
              ws[(long long)b * CHUNKS + 32 + lane];
    #pragma unroll
    for (int m = 16; m >= 1; m >>= 1) v += __shfl_xor(v, m, 32);
    if (lane == 0) out[b] = v;
}

extern "C" void kernel_launch(void* const* d_in, const int* in_sizes, int n_in,
                              void* d_out, int out_size, void* d_ws, size_t ws_size,
                              hipStream_t stream) {
    (void)n_in; (void)ws_size;
    const float* mean   = (const float*)d_in[0];
    const float* logvar = (const float*)d_in[1];
    const float* x      = (const float*)d_in[2];
    float* out = (float*)d_out;
    float* ws  = (float*)d_ws;   // needs out_size * CHUNKS floats (64*64*4 = 16 KB)

    const long long total       = (long long)in_sizes[0];       // 50,331,648
    const long long batches     = (long long)out_size;          // 64
    const long long n_per_batch = total / batches;              // 786,432
    const long long n_f4        = n_per_batch / 4;              // 196,608

    dim3 grid((unsigned)CHUNKS, (unsigned)batches);
    dg_partial_kernel<<<grid, TPB, 0, stream>>>(mean, logvar, x, ws, n_f4);
    dg_final_kernel<<<(unsigned)batches, 32, 0, stream>>>(ws, out);
}